// DeepFM_63909113365295
// MI455X (gfx1250) — compile-verified
//
#include <hip/hip_runtime.h>
#include <hip/hip_bf16.h>

typedef __attribute__((ext_vector_type(2))) float v2f;
typedef __attribute__((ext_vector_type(8))) float v8f;
typedef __attribute__((ext_vector_type(4))) int   v4i;

#define BATCH  4096
#define FIELDS 39
#define KDIM   16
#define FK     (FIELDS * KDIM)   // 624
#define TILE   32                // batch rows per block
#define HSTR   628               // padded LDS row stride (floats)
#define NBLK   (BATCH / TILE)    // 128

#if defined(__gfx1250__) && __has_builtin(__builtin_amdgcn_global_load_async_to_lds_b128)
#define USE_ASYNC_LDS 1
#else
#define USE_ASYNC_LDS 0
#endif

// pointer-to-(AS1 v4i) and pointer-to-(AS3 v4i)
typedef __attribute__((address_space(1))) v4i* g_v4i_p;
typedef __attribute__((address_space(3))) v4i* l_v4i_p;

__global__ __launch_bounds__(256)
void deepfm_fused_kernel(const int*   __restrict__ feat,     // [B, F]
                         const float* __restrict__ w_table,  // [FEA, 1]
                         const float* __restrict__ v_table,  // [FEA, K]
                         const float* __restrict__ w0,       // [1]
                         const float* __restrict__ W0,       // [624, 10]
                         const float* __restrict__ b0,       // [10]
                         const float* __restrict__ W1,       // [10, 5]
                         const float* __restrict__ b1,       // [5]
                         const float* __restrict__ W2,       // [5, 3]
                         const float* __restrict__ b2,       // [3]
                         const float* __restrict__ W3,       // [3, 1]
                         const float* __restrict__ b3,       // [1]
                         float*       __restrict__ out)      // [B, 1]
{
    __shared__ float hs[TILE * HSTR];   // gathered embeddings, 32 rows x 624 (+pad)
    __shared__ float w0s[FK * 16];      // W0 zero-padded 624 x 16
    __shared__ float h1s[TILE * 16];    // layer-0 output tile
    __shared__ float fm_lds[TILE];      // per-row FM term

    const int tid  = threadIdx.x;
    const int row0 = blockIdx.x * TILE;

    // ---- stage W0 (zero-padded to 16 cols) into LDS ----
    for (int i = tid; i < FK * 16; i += 256) {
        const int kk = i >> 4, n = i & 15;
        w0s[i] = (n < 10) ? W0[kk * 10 + n] : 0.0f;
    }

    // ---- gather embeddings: 4 lanes x 16B per 64B embedding row ----
    for (int t = tid; t < TILE * FIELDS * 4; t += 256) {
        const int g = t >> 2, q = t & 3;
        const int r = g / FIELDS, f = g % FIELDS;
        const int idx = feat[(row0 + r) * FIELDS + f];
        const float* src = v_table + (long long)idx * KDIM + q * 4;
        float* dst = &hs[r * HSTR + f * KDIM + q * 4];
#if USE_ASYNC_LDS
        // direct memory -> LDS async copy (GLOBAL_LOAD_ASYNC_TO_LDS_B128)
        __builtin_amdgcn_global_load_async_to_lds_b128(
            (g_v4i_p)(uintptr_t)src,
            (l_v4i_p)(unsigned)(uintptr_t)(void*)dst,
            0, 0);
#else
        const float4 v = *reinterpret_cast<const float4*>(src);
        dst[0] = v.x; dst[1] = v.y; dst[2] = v.z; dst[3] = v.w;
#endif
    }
#if USE_ASYNC_LDS
#if __has_builtin(__builtin_amdgcn_s_wait_asynccnt)
    __builtin_amdgcn_s_wait_asynccnt(0);
#else
    asm volatile("s_wait_asynccnt 0x0" ::: "memory");
#endif
#endif
    __syncthreads();

    // ---- FM closed form, parallel over all waves: 512 (row,k) tasks ----
    // task t: r = t>>4, k = t&15; lanes within a 16-lane group share a row.
    for (int t = tid; t < TILE * KDIM; t += 256) {
        const int r = t >> 4, k = t & 15;
        const float* hr = &hs[r * HSTR];
        float sk = 0.0f, ssq = 0.0f;
        for (int f = 0; f < FIELDS; ++f) {
            const float v = hr[f * KDIM + k];
            sk += v;
            ssq = fmaf(v, v, ssq);
        }
        float val = fmaf(sk, sk, -ssq);   // s_k^2 - sum_f v^2
        // deterministic tree reduction over the 16-lane group
        #pragma unroll
        for (int m = 8; m >= 1; m >>= 1)
            val += __shfl_xor(val, m, 32);
        if (k == 0) fm_lds[r] = 0.5f * val;
    }

    // ---- layer 0 GEMM via V_WMMA_F32_16X16X4_F32: waves 0..1, 16 rows each ----
    const int wave = tid >> 5;
    const int lane = tid & 31;
    if (wave < TILE / 16) {
        const int mbase = wave * 16;
        const int m     = lane & 15;          // A: row M / B: col N
        const int koff  = (lane >> 4) * 2;    // lanes 16-31 hold K+2, K+3
        v8f c = {};
        const float* arow = &hs[(mbase + m) * HSTR];
        for (int kk = 0; kk < FK; kk += 4) {
            v2f a, b;
            a.x = arow[kk + koff];
            a.y = arow[kk + koff + 1];
            b.x = w0s[(kk + koff) * 16 + m];
            b.y = w0s[(kk + koff + 1) * 16 + m];
            c = __builtin_amdgcn_wmma_f32_16x16x4_f32(
                    false, a, false, b, (short)0, c, false, false);
        }
        // C/D layout: VGPR j -> M = j (lanes 0-15) / j+8 (lanes 16-31); N = lane&15
        const int mo = (lane >> 4) * 8;
        #pragma unroll
        for (int j = 0; j < 8; ++j)
            h1s[(mbase + j + mo) * 16 + m] = c[j];
    }
    __syncthreads();

    // ---- linear term + tiny MLP tail per row ----
    if (tid < TILE) {
        const int r = tid;
        float lin = 0.0f;
        for (int f = 0; f < FIELDS; ++f)
            lin += w_table[feat[(row0 + r) * FIELDS + f]];
        float h1[10];
        #pragma unroll
        for (int n = 0; n < 10; ++n)
            h1[n] = fmaxf(h1s[r * 16 + n] + b0[n], 0.0f);
        float h2[5];
        #pragma unroll
        for (int n = 0; n < 5; ++n) {
            float acc = b1[n];
            #pragma unroll
            for (int k2 = 0; k2 < 10; ++k2) acc = fmaf(h1[k2], W1[k2 * 5 + n], acc);
            h2[n] = fmaxf(acc, 0.0f);
        }
        float h3[3];
        #pragma unroll
        for (int n = 0; n < 3; ++n) {
            float acc = b2[n];
            #pragma unroll
            for (int k2 = 0; k2 < 5; ++k2) acc = fmaf(h2[k2], W2[k2 * 3 + n], acc);
            h3[n] = fmaxf(acc, 0.0f);
        }
        float dnn = b3[0];
        #pragma unroll
        for (int k2 = 0; k2 < 3; ++k2) dnn = fmaf(h3[k2], W3[k2], dnn);
        out[row0 + r] = fm_lds[r] + lin + w0[0] + dnn;
    }
}

extern "C" void kernel_launch(void* const* d_in, const int* in_sizes, int n_in,
                              void* d_out, int out_size, void* d_ws, size_t ws_size,
                              hipStream_t stream) {
    (void)in_sizes; (void)n_in; (void)out_size; (void)d_ws; (void)ws_size;
    const int*   feat    = (const int*)  d_in[0];
    const float* w_table = (const float*)d_in[1];
    const float* v_table = (const float*)d_in[2];
    const float* w0      = (const float*)d_in[3];
    const float* W0      = (const float*)d_in[4];
    const float* b0      = (const float*)d_in[5];
    const float* W1      = (const float*)d_in[6];
    const float* b1      = (const float*)d_in[7];
    const float* W2      = (const float*)d_in[8];
    const float* b2      = (const float*)d_in[9];
    const float* W3      = (const float*)d_in[10];
    const float* b3      = (const float*)d_in[11];
    float* out = (float*)d_out;

    deepfm_fused_kernel<<<dim3(NBLK), dim3(256), 0, stream>>>(
        feat, w_table, v_table, w0, W0, b0, W1, b1, W2, b2, W3, b3, out);
}